// GCN_81338090651993
// MI455X (gfx1250) — compile-verified
//
#include <hip/hip_runtime.h>
#include <math.h>

typedef float v2f __attribute__((ext_vector_type(2)));
typedef float v8f __attribute__((ext_vector_type(8)));

#define INC   128
#define HID   96
#define OUTC  40
#define OUTCP 48   // OUTC padded to column-tile multiple
#define MASKC 64

// Hardware f32 atomic add, no return (STOREcnt-tracked). Avoids any CAS-loop
// lowering of atomicAdd(float*). L2-scope RMW is what we want: the whole
// aggregation working set is L2-resident (192 MB).
__device__ __forceinline__ void atomic_add_f32(float* p, float v) {
    asm volatile("global_atomic_add_f32 %0, %1, off" :: "v"(p), "v"(v) : "memory");
}

// ---------------------------------------------------------------- utilities
__global__ void zero_f32(float* __restrict__ p, long n) {
    long i = (long)blockIdx.x * blockDim.x + threadIdx.x;
    long stride = (long)gridDim.x * blockDim.x;
    for (; i < n; i += stride) p[i] = 0.0f;
}

__global__ void deg_kernel(const int* __restrict__ dst, float* __restrict__ deg, int E) {
    int i = blockIdx.x * blockDim.x + threadIdx.x;
    if (i < E) atomic_add_f32(&deg[dst[i]], 1.0f);
}

__global__ void dinv_kernel(const float* __restrict__ deg, float* __restrict__ dinv, int n) {
    int i = blockIdx.x * blockDim.x + threadIdx.x;
    if (i < n) dinv[i] = rsqrtf(deg[i] + 1.0f);   // +1 = self loop
}

// Zero-pad B [K,NB] -> Bp [K,NBP]
__global__ void pad_B(const float* __restrict__ W, float* __restrict__ Bp,
                      int K, int NB, int NBP) {
    int i = blockIdx.x * blockDim.x + threadIdx.x;
    int total = K * NBP;
    if (i >= total) return;
    int r = i / NBP, c = i - r * NBP;
    Bp[i] = (c < NB) ? W[r * NB + c] : 0.0f;
}

// ---------------------------------------------------------------- fp32 WMMA GEMM
// C[M,NB] = A[M,K] @ B[K,NBP] (+bias)(+relu). NBP is a multiple of 16 (B is
// pre-padded when NB isn't). One wave owns a 16-row strip and iterates over all
// NT = NBP/16 column tiles per K-step, so the A fragment is loaded once per
// K-step and reused NT times from registers. Accumulators stay independent so
// the NT back-to-back v_wmma ops pipeline without RAW hazard NOPs.
template<int K, int NB, int NBP, bool BIAS, bool RELU>
__global__ void wmma_gemm_f32(const float* __restrict__ A, const float* __restrict__ B,
                              const float* __restrict__ bias, float* __restrict__ C) {
    constexpr int NT = NBP / 16;
    const int lane    = threadIdx.x;                   // 0..31 (wave32)
    const int arow    = blockIdx.x * 16 + (lane & 15); // A row this lane feeds
    const int colbase = lane & 15;                     // column within tile
    const int kh      = (lane >> 4) * 2;               // K-pair select: 0 or 2

    v8f acc[NT] = {};
    for (int k = 0; k < K; k += 4) {
        // A 16x4 fragment: vreg r holds K = k+kh+r for row arow (b64 load)
        v2f a;
        a.x = A[(long)arow * K + k + kh];
        a.y = A[(long)arow * K + k + kh + 1];
#pragma unroll
        for (int nt = 0; nt < NT; ++nt) {
            // B 4x16 fragment: vreg r holds row K = k+kh+r, col nt*16+colbase
            v2f b;
            b.x = B[(long)(k + kh) * NBP + nt * 16 + colbase];
            b.y = B[(long)(k + kh + 1) * NBP + nt * 16 + colbase];
            // (neg_a, A, neg_b, B, c_mod, C, reuse_a, reuse_b)
            acc[nt] = __builtin_amdgcn_wmma_f32_16x16x4_f32(
                false, a, false, b, (short)0, acc[nt], false, false);
        }
    }

    // C/D layout: vreg r -> M = r + 8*(lane/16), N = colbase
    const int r0 = blockIdx.x * 16 + (lane >> 4) * 8;
#pragma unroll
    for (int nt = 0; nt < NT; ++nt) {
        const int col = nt * 16 + colbase;
        const bool colOK = (NB % 16 == 0) ? true : (col < NB);
        if (colOK) {
#pragma unroll
            for (int r = 0; r < 8; ++r) {
                float v = acc[nt][r];
                if constexpr (BIAS) v += bias[col];
                if constexpr (RELU) v = fmaxf(v, 0.0f);
                C[(long)(r0 + r) * NB + col] = v;
            }
        }
    }
}

// ---------------------------------------------------------------- edge scatter
// One wave per edge; lanes stripe the F features. Y[dst] += X[src] * dinv[s]*dinv[d]
template<int F>
__global__ void scatter_edges(const int* __restrict__ src, const int* __restrict__ dst,
                              const float* __restrict__ dinv, const float* __restrict__ X,
                              float* __restrict__ Y, int E) {
    const int lane = threadIdx.x & 31;
    const int warp = threadIdx.x >> 5;
    const int e = blockIdx.x * (blockDim.x >> 5) + warp;
    if (e >= E) return;
    const int s = src[e], d = dst[e];
    const float w = dinv[s] * dinv[d];
    const float* xs = X + (long)s * F;
    float* yd = Y + (long)d * F;
#pragma unroll
    for (int j = lane; j < F; j += 32)
        atomic_add_f32(&yd[j], xs[j] * w);
}

// h = relu(h_agg + h_lin * dinv^2 (self loop) + b1)
__global__ void finalize_h(const float* __restrict__ hagg, const float* __restrict__ hlin,
                           const float* __restrict__ dinv, const float* __restrict__ b1,
                           float* __restrict__ h, int n) {
    long i = (long)blockIdx.x * blockDim.x + threadIdx.x;
    long total = (long)n * HID;
    if (i >= total) return;
    int node = (int)(i / HID);
    int j = (int)(i - (long)node * HID);
    float di = dinv[node];
    float v = hagg[i] + hlin[i] * di * di + b1[j];
    h[i] = fmaxf(v, 0.0f);
}

// z = z_agg + z_lin * dinv^2 + b2; logits = log_softmax(z)
__global__ void finalize_z_logsoftmax(const float* __restrict__ zagg,
                                      const float* __restrict__ zlin,
                                      const float* __restrict__ dinv,
                                      const float* __restrict__ b2,
                                      float* __restrict__ logits, int n) {
    int i = blockIdx.x * blockDim.x + threadIdx.x;
    if (i >= n) return;
    float di2 = dinv[i] * dinv[i];
    float zr[OUTC];
    float mx = -INFINITY;
#pragma unroll
    for (int j = 0; j < OUTC; ++j) {
        float v = zagg[(long)i * OUTC + j] + zlin[(long)i * OUTC + j] * di2 + b2[j];
        zr[j] = v;
        mx = fmaxf(mx, v);
    }
    float se = 0.0f;
#pragma unroll
    for (int j = 0; j < OUTC; ++j) se += expf(zr[j] - mx);
    float lse = mx + logf(se);
#pragma unroll
    for (int j = 0; j < OUTC; ++j) logits[(long)i * OUTC + j] = zr[j] - lse;
}

// prob = sigmoid(hm @ Wm2 + bm2); mask = (prob > 0.5) (straight-through fwd)
__global__ void mask_head(const float* __restrict__ hm, const float* __restrict__ Wm2,
                          const float* __restrict__ bm2, float* __restrict__ prob,
                          float* __restrict__ maskp, int n) {
    int i = blockIdx.x * blockDim.x + threadIdx.x;
    if (i >= n) return;
    float acc = bm2[0];
#pragma unroll
    for (int j = 0; j < MASKC; ++j) acc += hm[(long)i * MASKC + j] * Wm2[j];
    float p = 1.0f / (1.0f + expf(-acc));
    prob[i] = p;
    maskp[i] = (p > 0.5f) ? 1.0f : 0.0f;
}

// ---------------------------------------------------------------- launcher
extern "C" void kernel_launch(void* const* d_in, const int* in_sizes, int n_in,
                              void* d_out, int out_size, void* d_ws, size_t ws_size,
                              hipStream_t stream) {
    const float* x   = (const float*)d_in[0];
    const int*   ei  = (const int*)d_in[1];
    const float* W1  = (const float*)d_in[2];
    const float* b1  = (const float*)d_in[3];
    const float* W2  = (const float*)d_in[4];
    const float* b2  = (const float*)d_in[5];
    const float* Wm1 = (const float*)d_in[6];
    const float* bm1 = (const float*)d_in[7];
    const float* Wm2 = (const float*)d_in[8];
    const float* bm2 = (const float*)d_in[9];

    const int N = in_sizes[0] / INC;   // 50000 (exactly 3125*16)
    const int E = in_sizes[1] / 2;     // 800000
    const int* src = ei;
    const int* dst = ei + E;

    // workspace layout (floats): deg[N] dinv[N] bufA[96N] bufB[96N] h[96N] W2p[96*48]
    float* ws   = (float*)d_ws;
    float* deg  = ws;
    float* dinv = deg + N;
    float* bufA = dinv + N;                 // h_lin -> hm -> z_lin
    float* bufB = bufA + (long)N * HID;     // h_agg -> z_agg
    float* h    = bufB + (long)N * HID;
    float* W2p  = h + (long)N * HID;        // padded [HID, OUTCP]

    float* logits = (float*)d_out;          // [N, OUTC]
    float* prob   = logits + (long)N * OUTC;
    float* maskp  = prob + N;

    // 1) zero degree + h aggregation buffer; pad W2
    zero_f32<<<2048, 256, 0, stream>>>(deg, N);
    zero_f32<<<4096, 256, 0, stream>>>(bufB, (long)N * HID);
    pad_B<<<(HID * OUTCP + 255) / 256, 256, 0, stream>>>(W2, W2p, HID, OUTC, OUTCP);
    // 2) degree count, 3) dinv
    deg_kernel<<<(E + 255) / 256, 256, 0, stream>>>(dst, deg, E);
    dinv_kernel<<<(N + 255) / 256, 256, 0, stream>>>(deg, dinv, N);

    // 4) h_lin = x @ W1    (WMMA fp32, one wave per 16-row strip)
    wmma_gemm_f32<INC, HID, HID, false, false>
        <<<N / 16, 32, 0, stream>>>(x, W1, nullptr, bufA);
    // 5) scatter edges for layer 1; 6) finalize h (self loop + bias + relu)
    scatter_edges<HID><<<(E + 7) / 8, 256, 0, stream>>>(src, dst, dinv, bufA, bufB, E);
    finalize_h<<<(int)(((long)N * HID + 255) / 256), 256, 0, stream>>>(bufB, bufA, dinv, b1, h, N);

    // 7) hm = relu(h @ Wm1 + bm1)  (reuse bufA, stride MASKC); 8) mask head
    wmma_gemm_f32<HID, MASKC, MASKC, true, true>
        <<<N / 16, 32, 0, stream>>>(h, Wm1, bm1, bufA);
    mask_head<<<(N + 255) / 256, 256, 0, stream>>>(bufA, Wm2, bm2, prob, maskp, N);

    // 9) z_lin = h @ W2p (reuse bufA, stride OUTC; B pre-padded to OUTCP)
    wmma_gemm_f32<HID, OUTC, OUTCP, false, false>
        <<<N / 16, 32, 0, stream>>>(h, W2p, nullptr, bufA);
    // 10) zero z aggregation (reuse bufB); 11) scatter; 12) finalize + log_softmax
    zero_f32<<<4096, 256, 0, stream>>>(bufB, (long)N * OUTC);
    scatter_edges<OUTC><<<(E + 7) / 8, 256, 0, stream>>>(src, dst, dinv, bufA, bufB, E);
    finalize_z_logsoftmax<<<(N + 255) / 256, 256, 0, stream>>>(bufB, bufA, dinv, b2, logits, N);
}